// JointEntropy_70136815944311
// MI455X (gfx1250) — compile-verified
//
#include <hip/hip_runtime.h>

// JointEntropy (KDE patch entropy), R=3, h=0.5, x:[4,3,256,256] f32 -> [4,254,254] f32
//
// Per pixel: 9 patches x 3 channels. Gram = P P^T on V_WMMA_F32_16X16X4_F32;
// a second accumulating WMMA adds the rank-2 (-2 sq_n - 2 sq_m) term so the
// matrix pipe emits logits = -2*d^2 directly. Masked exp, then a merged
// multi-row ds_swizzle xor-reduction (15 swizzles for 16 row-sums), ONE log
// per lane, and a weighted 5-stage full-wave xor-sum for the final scalar.
// One wave per output pixel; problem is tiny (3MB in / 1MB out, L2-resident).

typedef __attribute__((ext_vector_type(2))) float v2f;
typedef __attribute__((ext_vector_type(8))) float v8f;

#define RS   3
#define NP   9          // patches per pixel (R*R)
#define HP   254        // H - R + 1
#define WPX  254        // W - R + 1
#define CHAN 3
#define HH   256
#define WW   256

// log(9) + 3*log(sqrt(2*pi)*0.5)
#define LOG_NORM 2.8745986366271973f

// ds_swizzle_b32 group-of-32 xor pattern: offset = {0, xor[4:0], or=0, and=0x1f}
template<int XORM>
__device__ __forceinline__ float swz_xor(float v) {
  return __int_as_float(__builtin_amdgcn_ds_swizzle(__float_as_int(v),
                                                    (XORM << 10) | 0x1f));
}

// One merge stage of the multi-row reduction: finish xor step BIT on rows a,b,
// then keep row a in lanes with (lane&BIT)==0 and row b in the others.
template<int BIT>
__device__ __forceinline__ float merge_pair(float a, float b, int lane) {
  const float as = a + swz_xor<BIT>(a);
  const float bs = b + swz_xor<BIT>(b);
  return (lane & BIT) ? bs : as;
}

__global__ __launch_bounds__(256)
void kde_entropy_wmma(const float* __restrict__ x, float* __restrict__ out, int total) {
  const int lane = threadIdx.x & 31;
  const int wave = blockIdx.x * (blockDim.x >> 5) + (threadIdx.x >> 5);
  if (wave >= total) return;               // whole wave exits together

  const int j = wave % WPX;
  const int i = (wave / WPX) % HP;
  const int b = wave / (WPX * HP);

  // Lane roles: lanes 0..8 load ch0/ch1 of patch n=lane; lanes 16..24 load ch2
  // of patch n=lane-16. Other lanes are zero padding (rows/cols 9..15, K=3).
  const int  n  = lane & 15;
  const bool lo = lane < 16;
  float ch0 = 0.f, ch1 = 0.f, ch2 = 0.f;
  if (n < NP) {
    const int di = n / RS, dj = n % RS;
    const size_t base = (size_t)b * (CHAN * HH * WW) + (size_t)(i + di) * WW + (size_t)(j + dj);
    if (lo) {
      ch0 = x[base];                         // channel 0
      ch1 = x[base + (size_t)(HH * WW)];     // channel 1
    } else {
      ch2 = x[base + (size_t)(2 * HH * WW)]; // channel 2
    }
  }

  // ||p_n||^2 in the low half (high half sq == 0 by construction)
  const float ch2x = swz_xor<16>(ch2);
  const float sq   = ch0 * ch0 + ch1 * ch1 + ch2x * ch2x;

  // WMMA #1: D1 = (2P)(2P)^T = 4*Gram.
  // A-16x4 layout: v0 = (lane<16 ? 2*ch0 : 2*ch2) -> K0/K2 ; v1 = 2*ch1 -> K1/K3(=0).
  // The 4x16 B layout demands the *same* per-lane contents -> pass A twice.
  v2f a;
  a.x = 2.f * (ch0 + ch2);                   // disjoint halves
  a.y = 2.f * ch1;                           // high half ch1==0 -> K3 = 0
  v8f acc = {};
  acc = __builtin_amdgcn_wmma_f32_16x16x4_f32(false, a, false, a,
                                              (short)0, acc, false, false);

  // WMMA #2: logits = 4*Gram - 2*sq_M - 2*sq_N  (rank-2 update, accumulate).
  const float lowone = lo ? 1.f : 0.f;
  v2f a2, b2;
  a2.x = -2.f * sq;  a2.y = lowone;          // A2[M,:] = [-2 sq_M, 1, 0, 0]
  b2.x = lowone;     b2.y = -2.f * sq;       // B2[:,N] = [1, -2 sq_N, 0, 0]^T
  acc = __builtin_amdgcn_wmma_f32_16x16x4_f32(false, a2, false, b2,
                                              (short)0, acc, false, false);

  // Masked exp. D layout: VGPR r, lane L -> M = r + 8*(L/16), N = L%16.
  // Fold the column mask (N>8 = padding) into the clamp bound: exp(-inf)=0.
  // Diagonal logit == 0, so no max-shift is needed for the logsumexp.
  const float bound = (n < NP) ? 0.f : -__builtin_inff();
  float e[8];
#pragma unroll
  for (int r = 0; r < 8; ++r)
    e[r] = __expf(fminf(acc[r], bound));

  // Merged multi-row xor-reduction within 16-lane halves:
  // after this, lane l holds S_{(l&7) + 8*(l>=16)}.
  float m0 = merge_pair<1>(e[0], e[1], lane);
  float m1 = merge_pair<1>(e[2], e[3], lane);
  float m2 = merge_pair<1>(e[4], e[5], lane);
  float m3 = merge_pair<1>(e[6], e[7], lane);
  float p0 = merge_pair<2>(m0, m1, lane);
  float p1 = merge_pair<2>(m2, m3, lane);
  float q  = merge_pair<4>(p0, p1, lane);
  q += swz_xor<8>(q);

  // One log per lane; weight so the full-wave sum counts rows 0..8 exactly once:
  // low half holds each of rows 0..7 twice (w=0.5); high lanes l&7==0 hold row 8
  // twice (w=0.5); high lanes l&7!=0 hold garbage rows 9..15 (w=0, logs finite).
  const float wgt = (lo || (lane & 7) == 0) ? 0.5f : 0.f;
  float w = __logf(q) * wgt;
  w += swz_xor<1>(w);
  w += swz_xor<2>(w);
  w += swz_xor<4>(w);
  w += swz_xor<8>(w);
  w += swz_xor<16>(w);

  if (lane == 0)
    out[wave] = LOG_NORM - w * (1.f / 9.f);
}

extern "C" void kernel_launch(void* const* d_in, const int* in_sizes, int n_in,
                              void* d_out, int out_size, void* d_ws, size_t ws_size,
                              hipStream_t stream) {
  (void)in_sizes; (void)n_in; (void)d_ws; (void)ws_size;
  const float* x   = (const float*)d_in[0];
  float*       out = (float*)d_out;
  const int total = out_size;                    // 4*254*254 = 258064 pixels
  const int wavesPerBlock = 256 / 32;            // 8 waves per block
  const int blocks = (total + wavesPerBlock - 1) / wavesPerBlock;
  kde_entropy_wmma<<<blocks, 256, 0, stream>>>(x, out, total);
}